// DRSolver_7232724926817
// MI455X (gfx1250) — compile-verified
//
#include <hip/hip_runtime.h>
#include <math.h>

// Problem constants (from reference)
#define XD 64
#define NEQ 16
#define NINEQ 32
#define NTOT 96           // X_DIM + N_INEQ
#define MCON 48           // N_EQ + N_INEQ
#define PDIM 112          // X_DIM + N_EQ + N_INEQ
#define BATCH 4096
#define NUM_STEPS 10
#define BOUND 1000.0f

typedef __attribute__((ext_vector_type(2))) float v2f;
typedef __attribute__((ext_vector_type(8))) float v8f;

// ---------------- workspace layout (floats) ----------------
// Ptp  : 48*96 float2  = 9216 floats   (rows k=2kk,2kk+1 of P^T interleaved)
// Wutp : 56*96 float2  = 10752 floats  (rows k of Wu^T, pair-interleaved)
// then precompute scratch.
#define WS_PTP   0
#define WS_WUTP  9216
#define WS_M1    19968          // 64*64
#define WS_E1    24064          // 64*48
#define WS_MINV  27136          // 64*64
#define WS_K     31232          // 48*48
#define WS_T     33536          // 48*96
#define WS_S2    38144          // 48*96
// total ~42752 floats = 171 KB of d_ws

__device__ void chol_inplace(float* Mat, int n, int tid, int nt) {
    for (int k = 0; k < n; ++k) {
        if (tid == 0) Mat[k*n+k] = sqrtf(Mat[k*n+k]);
        __syncthreads();
        float dk = Mat[k*n+k];
        for (int i = k+1+tid; i < n; i += nt) Mat[i*n+k] /= dk;
        __syncthreads();
        int rem = n-1-k;
        for (int t = tid; t < rem*rem; t += nt) {
            int i = k+1 + t/rem, j = k+1 + t%rem;
            if (j <= i) Mat[i*n+j] -= Mat[i*n+k]*Mat[j*n+k];
        }
        __syncthreads();
    }
}

// One-time factorization: builds P (96x96, symmetric) and Wu^T (112x96),
// stored in WMMA-B-fragment friendly pair-interleaved float2 layout.
__global__ void dr_precompute(const float* __restrict__ Q,
                              const float* __restrict__ A,
                              const float* __restrict__ G,
                              float* __restrict__ ws) {
    const int tid = threadIdx.x, nt = blockDim.x;
    float* M1   = ws + WS_M1;
    float* E1   = ws + WS_E1;
    float* Minv = ws + WS_MINV;
    float* K    = ws + WS_K;
    float* T    = ws + WS_T;
    float* S2   = ws + WS_S2;

    // M1 = Q + I  (Md upper-left block; GAMMA/2 == 1)
    for (int i = tid; i < XD*XD; i += nt)
        M1[i] = Q[i] + ((i/XD) == (i%XD) ? 1.f : 0.f);
    __syncthreads();
    chol_inplace(M1, XD, tid, nt);            // L1 (lower) in place

    // Solve L1 L1^T * col = rhs for 112 columns:
    //   cols 0..47  : rhs = [A^T | G^T]  -> E1 (64x48)
    //   cols 48..111: rhs = I            -> Minv11 (64x64)
    if (tid < MCON + XD) {
        float col[XD];
        if (tid < MCON) {
            for (int i = 0; i < XD; ++i)
                col[i] = (tid < NEQ) ? A[tid*XD + i] : G[(tid-NEQ)*XD + i];
        } else {
            int j = tid - MCON;
            for (int i = 0; i < XD; ++i) col[i] = (i == j) ? 1.f : 0.f;
        }
        for (int i = 0; i < XD; ++i) {                    // forward
            float s = col[i];
            for (int k = 0; k < i; ++k) s -= M1[i*XD+k]*col[k];
            col[i] = s / M1[i*XD+i];
        }
        for (int i = XD-1; i >= 0; --i) {                 // backward
            float s = col[i];
            for (int k = i+1; k < XD; ++k) s -= M1[k*XD+i]*col[k];
            col[i] = s / M1[i*XD+i];
        }
        if (tid < MCON) for (int i = 0; i < XD; ++i) E1[i*MCON+tid] = col[i];
        else            for (int i = 0; i < XD; ++i) Minv[i*XD + (tid-MCON)] = col[i];
    }
    __syncthreads();

    // K = J * Minv * J^T = [A;G] @ E1 + [[0,0],[0,I]]
    for (int idx = tid; idx < MCON*MCON; idx += nt) {
        int r = idx/MCON, c = idx%MCON;
        const float* Jr = (r < NEQ) ? (A + r*XD) : (G + (r-NEQ)*XD);
        float s = 0.f;
        for (int i = 0; i < XD; ++i) s += Jr[i]*E1[i*MCON+c];
        if (r >= NEQ && c == r) s += 1.f;
        K[idx] = s;
    }
    __syncthreads();
    chol_inplace(K, MCON, tid, nt);           // Lk (lower) in place

    // T = Lk^{-1} E^T (48x96); S2 = Lk^{-T} T (48x96). Thread per column c.
    if (tid < NTOT) {
        float t[MCON];
        for (int r = 0; r < MCON; ++r) {
            float e = (tid < XD) ? E1[tid*MCON + r]
                                 : ((r >= NEQ && (tid-XD) == (r-NEQ)) ? 1.f : 0.f);
            for (int j = 0; j < r; ++j) e -= K[r*MCON+j]*t[j];
            t[r] = e / K[r*MCON+r];
            T[r*NTOT+tid] = t[r];
        }
        for (int r = MCON-1; r >= 0; --r) {
            float s = t[r];
            for (int j = r+1; j < MCON; ++j) s -= K[j*MCON+r]*t[j];
            t[r] = s / K[r*MCON+r];
            S2[r*NTOT+tid] = t[r];
        }
    }
    __syncthreads();

    // P = Minv_full - T^T T  (symmetric). Store P^T == P pair-interleaved.
    float* Ptp  = ws + WS_PTP;
    float* Wutp = ws + WS_WUTP;
    for (int idx = tid; idx < NTOT*NTOT; idx += nt) {
        int k = idx/NTOT, n = idx%NTOT;
        float mv = (k < XD && n < XD) ? Minv[k*XD+n] : ((k == n) ? 1.f : 0.f);
        float s = 0.f;
        for (int r = 0; r < MCON; ++r) s += T[r*NTOT+k]*T[r*NTOT+n];
        Ptp[((k>>1)*NTOT + n)*2 + (k&1)] = mv - s;
    }
    __syncthreads();
    // Wu^T rows: k<64 -> -P[k][n] (symmetry), k>=64 -> S2[k-64][n]
    for (int idx = tid; idx < PDIM*NTOT; idx += nt) {
        int k = idx/NTOT, n = idx%NTOT;
        float wv = (k < XD) ? -Ptp[((k>>1)*NTOT + n)*2 + (k&1)]
                            : S2[(k-XD)*NTOT + n];
        Wutp[((k>>1)*NTOT + n)*2 + (k&1)] = wv;
    }
}

// ---------------- batched DR iteration kernel ----------------
// 64 threads (2 waves), 16 batch rows per block, 256 blocks.
// Wave w owns output columns [w*48, w*48+48) as three 16-col WMMA tiles.
#define STG_STRIDE 116   // float stride: 116 mod 64 = 52, gcd 4 -> conflict-free column reads
#define W2_STRIDE  112   // float2 stride: 112 mod 32 = 16 -> halves hit disjoint bank sets

__global__ __launch_bounds__(64)
void dr_solver(const float* __restrict__ x_in,
               const float* __restrict__ parms,
               const float2* __restrict__ Ptp,
               const float2* __restrict__ Wutp,
               float* __restrict__ out) {
    __shared__ float2 sW[56 * W2_STRIDE];     // 50176 B: Wu^T then P^T, K-pair rows
    __shared__ float  stg[16 * STG_STRIDE];   // 7424 B: parms tile, then x tile

    const int tid  = threadIdx.x;
    const int lane = tid & 31;
    const int w    = tid >> 5;        // wave 0/1
    const int m    = lane & 15;       // A: row-in-tile; B: col-in-tile
    const int kh   = lane >> 4;       // half-wave selects K offset (+2)
    const int row0 = blockIdx.x * 16;

    // Phase 0: Wu^T -> LDS, parms tile -> LDS
    for (int idx = tid; idx < 56*NTOT; idx += 64)
        sW[(idx/NTOT)*W2_STRIDE + (idx%NTOT)] = Wutp[idx];
    for (int idx = tid; idx < 16*PDIM; idx += 64)
        stg[(idx/PDIM)*STG_STRIDE + (idx%PDIM)] = parms[(row0 + idx/PDIM)*PDIM + (idx%PDIM)];
    __syncthreads();

    // u fragments: u = Wu @ parms_row, accumulated over K=112 in steps of 4
    v8f U[3];
    #pragma unroll
    for (int t = 0; t < 3; ++t) {
        const int n0 = w*48 + t*16;
        v8f c = {0.f,0.f,0.f,0.f,0.f,0.f,0.f,0.f};
        #pragma unroll
        for (int k0 = 0; k0 < PDIM; k0 += 4) {
            v2f a = *(const v2f*)&stg[m*STG_STRIDE + k0 + kh*2];
            v2f b = *(const v2f*)&sW[((k0>>1) + kh)*W2_STRIDE + n0 + m];
            c = __builtin_amdgcn_wmma_f32_16x16x4_f32(false, a, false, b,
                                                      (short)0, c, false, false);
        }
        U[t] = c;
    }
    __syncthreads();

    // Phase 1: P^T -> LDS (over Wu^T), x tile -> LDS
    for (int idx = tid; idx < 48*NTOT; idx += 64)
        sW[(idx/NTOT)*W2_STRIDE + (idx%NTOT)] = Ptp[idx];
    for (int idx = tid; idx < 16*NTOT; idx += 64)
        stg[(idx/NTOT)*STG_STRIDE + (idx%NTOT)] = x_in[(row0 + idx/NTOT)*NTOT + (idx%NTOT)];
    __syncthreads();

    v8f Y[3];
    for (int it = 0; it < NUM_STEPS; ++it) {
        // y = P x + u : 3 independent WMMA chains per wave (hides RAW latency)
        #pragma unroll
        for (int t = 0; t < 3; ++t) {
            const int n0 = w*48 + t*16;
            v8f c = U[t];
            #pragma unroll
            for (int k0 = 0; k0 < NTOT; k0 += 4) {
                v2f a = *(const v2f*)&stg[m*STG_STRIDE + k0 + kh*2];
                v2f b = *(const v2f*)&sW[((k0>>1) + kh)*W2_STRIDE + n0 + m];
                c = __builtin_amdgcn_wmma_f32_16x16x4_f32(false, a, false, b,
                                                          (short)0, c, false, false);
            }
            Y[t] = c;
        }
        __syncthreads();   // all waves done reading x fragments
        // z = clamp(2y - x, l, 1000); x += z - y   (done in D-fragment layout)
        #pragma unroll
        for (int t = 0; t < 3; ++t) {
            const int n0 = w*48 + t*16;
            #pragma unroll
            for (int r = 0; r < 8; ++r) {
                const int row = r + kh*8;
                const int col = n0 + m;
                float xv = stg[row*STG_STRIDE + col];
                float yv = Y[t][r];
                float v  = 2.f*yv - xv;
                float lo = (col < XD) ? -BOUND : 0.f;
                float z  = fminf(fmaxf(v, lo), BOUND);
                stg[row*STG_STRIDE + col] = xv + z - yv;
            }
        }
        __syncthreads();   // x tile updated before next iteration's reads
    }

    // Output: last y, columns [0,64)
    #pragma unroll
    for (int t = 0; t < 3; ++t) {
        const int n0 = w*48 + t*16;
        if (n0 < XD) {
            #pragma unroll
            for (int r = 0; r < 8; ++r) {
                const int row = r + kh*8;
                out[(row0 + row)*XD + n0 + m] = Y[t][r];
            }
        }
    }
}

extern "C" void kernel_launch(void* const* d_in, const int* in_sizes, int n_in,
                              void* d_out, int out_size, void* d_ws, size_t ws_size,
                              hipStream_t stream) {
    const float* x     = (const float*)d_in[0];   // (4096, 96)
    const float* parms = (const float*)d_in[1];   // (4096, 112)
    const float* Q     = (const float*)d_in[2];   // (64, 64)
    const float* A     = (const float*)d_in[3];   // (16, 64)
    const float* G     = (const float*)d_in[4];   // (32, 64)
    float* ws = (float*)d_ws;

    dr_precompute<<<1, 128, 0, stream>>>(Q, A, G, ws);
    dr_solver<<<BATCH/16, 64, 0, stream>>>(x, parms,
                                           (const float2*)(ws + WS_PTP),
                                           (const float2*)(ws + WS_WUTP),
                                           (float*)d_out);
}